// KDEGaussian_16673063043687
// MI455X (gfx1250) — compile-verified
//
#include <hip/hip_runtime.h>
#include <math.h>

typedef __attribute__((ext_vector_type(2))) float v2f;
typedef __attribute__((ext_vector_type(8))) float v8f;

#define KDE_B 4
#define KDE_N 4096
#define KDE_M (128 * 128)
#define KDE_BW 0.1f

// One wave owns a 16-location tile of one batch and reduces over all N samples.
// Exponent computed entirely inside V_WMMA_F32_16X16X4_F32 via K-augmentation:
//   A[n,:] = (x, y, ||s||^2, 1)
//   B[:,m] = (2c*lx, 2c*ly, -c, -c*||l||^2),  c = inv * log2(e)
// => D[n,m] = -c * d2(n,m) = log2(gaussian weight). Clamp to <=0, exp2, sum.
__global__ __launch_bounds__(256) void kde_tiles_kernel(
    const float* __restrict__ samples,   // (B, N, 2)
    const float* __restrict__ locs,      // (M, 2)
    float* __restrict__ out) {           // (B, M) unnormalized
  const float inv = 1.0f / (2.0f * KDE_BW * KDE_BW);       // 50
  const float c   = inv * 1.44269504088896340736f;         // inv * log2(e)

  const int tid   = threadIdx.x;
  const int lane  = tid & 31;
  const int wave  = tid >> 5;
  const int gid   = blockIdx.x * (blockDim.x >> 5) + wave; // 0..4095
  const int b     = gid >> 10;                             // 0..3
  const int m0    = (gid & 1023) << 4;                     // location tile base
  const int hi    = lane >> 4;                             // 0: K=0,1  1: K=2,3
  const int sub   = lane & 15;

  // B operand (4x16), loop-invariant. VGPR0 = K{0|2}, VGPR1 = K{1|3}.
  const float2 lv = ((const float2*)locs)[m0 + sub];
  const float l2  = lv.x * lv.x + lv.y * lv.y;
  v2f bm;
  bm.x = hi ? -c      : 2.0f * c * lv.x;
  bm.y = hi ? -c * l2 : 2.0f * c * lv.y;

  const float2* sp = (const float2*)samples + (size_t)b * KDE_N;

  v8f acc = {0.f, 0.f, 0.f, 0.f, 0.f, 0.f, 0.f, 0.f};

#pragma unroll 4
  for (int n0 = 0; n0 < KDE_N; n0 += 16) {
    const float2 sv = sp[n0 + sub];
    const float  s2 = sv.x * sv.x + sv.y * sv.y;
    v2f am;
    am.x = hi ? s2   : sv.x;   // K=0 -> x,  K=2 -> ||s||^2
    am.y = hi ? 1.0f : sv.y;   // K=1 -> y,  K=3 -> 1
    v8f cz = {0.f, 0.f, 0.f, 0.f, 0.f, 0.f, 0.f, 0.f};
    // D element (vgpr v, lane l): sample row v + 8*hi, location col l&15.
    v8f d = __builtin_amdgcn_wmma_f32_16x16x4_f32(
        /*neg_a=*/false, am, /*neg_b=*/false, bm,
        /*c_mod=*/(short)0, cz, /*reuse_a=*/false, /*reuse_b=*/false);
#pragma unroll
    for (int v = 0; v < 8; ++v) {
      // min(d, 0) in one v_med3_f32 (no canonicalize), == max(d2,0) clamp.
      float e = __builtin_amdgcn_fmed3f(d[v], -INFINITY, 0.0f);
      acc[v] += __builtin_amdgcn_exp2f(e);         // v_exp_f32
    }
  }

  // Reduce 8 accumulators (8 sample-rows) then the two lane halves (other 8).
  float s = acc[0] + acc[1] + acc[2] + acc[3] + acc[4] + acc[5] + acc[6] + acc[7];
  s += __shfl_xor(s, 16, 32);
  if (lane < 16) out[(size_t)b * KDE_M + m0 + lane] = s;
}

// Per-batch normalization, in place. One block per batch.
__global__ __launch_bounds__(256) void kde_norm_kernel(float* __restrict__ out) {
  __shared__ float red[256];
  float* p = out + (size_t)blockIdx.x * KDE_M;

  float local = 0.0f;
  for (int i = threadIdx.x; i < KDE_M; i += 256) local += p[i];
  red[threadIdx.x] = local;
  __syncthreads();
  for (int st = 128; st > 0; st >>= 1) {
    if (threadIdx.x < st) red[threadIdx.x] += red[threadIdx.x + st];
    __syncthreads();
  }
  const float invn = 1.0f / red[0];
  for (int i = threadIdx.x; i < KDE_M; i += 256) p[i] *= invn;
}

extern "C" void kernel_launch(void* const* d_in, const int* in_sizes, int n_in,
                              void* d_out, int out_size, void* d_ws, size_t ws_size,
                              hipStream_t stream) {
  const float* samples = (const float*)d_in[0];  // (4, 4096, 2) f32
  const float* locs    = (const float*)d_in[1];  // (128, 128, 2) f32
  float* out = (float*)d_out;                    // (4, 128, 128) f32

  // 4096 waves total = B(4) * M-tiles(1024); 8 waves per 256-thread block.
  kde_tiles_kernel<<<512, 256, 0, stream>>>(samples, locs, out);
  kde_norm_kernel<<<KDE_B, 256, 0, stream>>>(out);
}